// UpConvCropCat_44349832298548
// MI455X (gfx1250) — compile-verified
//
#include <hip/hip_runtime.h>

// ---------------------------------------------------------------------------
// UpConvCropCat for MI455X (gfx1250, wave32)
//   out[0:64]   = WMMA f32 channel-mix of nearest-up(x) + bias  (computed on
//                 pre-upsample grid, scattered 2x2x2)
//   out[64:128] = center crop of z
// ---------------------------------------------------------------------------

typedef __attribute__((ext_vector_type(2))) float v2f;
typedef __attribute__((ext_vector_type(8))) float v8f;

#define XS      131072          // 32*64*64 spatial positions of x
#define OD      64
#define OH      128
#define OWD     128
#define ODHW    (OD*OH*OWD)     // 1048576 per output channel
#define ZDD     68
#define ZHH     132
#define ZWW     132

// One wave handles a 16-position spatial tile (m0..m0+15) for ALL 64 output
// channels: 4 accumulators of 16x16, K=64 in 16 steps of V_WMMA_F32_16X16X4_F32.
__global__ __launch_bounds__(256) void upconv_wmma_kernel(
    const float* __restrict__ x, const float* __restrict__ Wm,
    const float* __restrict__ bias, float* __restrict__ out)
{
    __shared__ float lds_w[64 * 65];   // padded rows: kills bank conflicts
    __shared__ float lds_b[64];

    // cooperative stage of W (64x64 fp32, 16 KiB) into LDS, coalesced
    for (int i = threadIdx.x; i < 64 * 64; i += 256) {
        lds_w[(i >> 6) * 65 + (i & 63)] = Wm[i];
    }
    if (threadIdx.x < 64) lds_b[threadIdx.x] = bias[threadIdx.x];
    __syncthreads();

    const int lane  = threadIdx.x & 31;
    const int wave  = threadIdx.x >> 5;
    const int m0    = (blockIdx.x * 8 + wave) * 16;   // 8192 waves cover XS
    const int lm    = lane & 15;
    const int khalf = (lane >> 4) << 1;               // lanes 0-15: K+{0,1}; 16-31: K+{2,3}

    v8f acc[4] = {v8f{}, v8f{}, v8f{}, v8f{}};
    const float* xp = x + m0 + lm;

    #pragma unroll
    for (int k = 0; k < 64; k += 4) {
        const int kb = k + khalf;
        // B tile (4x16, K x spatial): contiguous 64B per half-wave, coalesced
        v2f bv;
        bv.x = xp[(size_t)(kb + 0) * XS];
        bv.y = xp[(size_t)(kb + 1) * XS];
        #pragma unroll
        for (int t = 0; t < 4; ++t) {
            // A tile (16x4, o x K) from padded LDS
            const int row = t * 16 + lm;
            v2f av;
            av.x = lds_w[row * 65 + kb];
            av.y = lds_w[row * 65 + kb + 1];
            acc[t] = __builtin_amdgcn_wmma_f32_16x16x4_f32(
                false, av, false, bv, (short)0, acc[t], false, false);
        }
    }

    // D layout: acc[t][j] -> o = 16t + j + 8*(lane>=16), m = m0 + (lane&15)
    const int m  = m0 + lm;
    const int zw = m & 63;
    const int zh = (m >> 6) & 63;
    const int zd = m >> 12;
    const size_t sp = (size_t)(zd * 2) * (OH * OWD) + (size_t)(zh * 2) * OWD + (zw * 2);
    const int ohalf = (lane >> 4) << 3;

    #pragma unroll
    for (int t = 0; t < 4; ++t) {
        #pragma unroll
        for (int j = 0; j < 8; ++j) {
            const int o = t * 16 + j + ohalf;
            const float v = acc[t][j] + lds_b[o];
            const float2 vv = make_float2(v, v);
            float* p = out + (size_t)o * ODHW + sp;
            // 2x2x2 nearest-neighbor scatter; each half-wave store = 128B contiguous
            *(float2*)(p)                 = vv;
            *(float2*)(p + OWD)           = vv;
            *(float2*)(p + OH * OWD)      = vv;
            *(float2*)(p + OH * OWD + OWD) = vv;
        }
    }
}

// Center-crop copy: out[64+c, d, h, :] = z[c, d+2, h+2, 2:130].
// z row offsets are ≡ 2 (mod 4) floats -> float2 is the widest aligned vector.
__global__ __launch_bounds__(256) void cropcat_kernel(
    const float* __restrict__ z, float* __restrict__ out)
{
    const int tid = blockIdx.x * 256 + threadIdx.x;   // 2^25 threads
    const int w2 = tid & 63;            // 64 float2 per row
    const int h  = (tid >> 6) & 127;
    const int d  = (tid >> 13) & 63;
    const int c  = tid >> 19;
    const size_t zi = (((size_t)c * ZDD + (d + 2)) * ZHH + (h + 2)) * (size_t)ZWW + 2 + (w2 << 1);
    const size_t oi = (((size_t)(64 + c) * OD + d) * OH + h) * (size_t)OWD + (w2 << 1);
    *(float2*)(out + oi) = *(const float2*)(z + zi);
}

extern "C" void kernel_launch(void* const* d_in, const int* in_sizes, int n_in,
                              void* d_out, int out_size, void* d_ws, size_t ws_size,
                              hipStream_t stream) {
    const float* x  = (const float*)d_in[0];
    const float* z  = (const float*)d_in[1];
    const float* Wm = (const float*)d_in[2];
    const float* b  = (const float*)d_in[3];
    float* out = (float*)d_out;

    // 8192 waves (16 spatial positions each) = 1024 blocks of 8 waves
    upconv_wmma_kernel<<<1024, 256, 0, stream>>>(x, Wm, b, out);
    // 64*64*128*64 float2 copies = 131072 blocks of 256 threads
    cropcat_kernel<<<131072, 256, 0, stream>>>(z, out);
}